// ContrastiveAutoencoder_16037407883756
// MI455X (gfx1250) — compile-verified
//
#include <hip/hip_runtime.h>
#include <math.h>

#define NN 100000
#define NL 100000
#define NE 1600000
#define DN 200
#define DL 300
#define CC 128

typedef float v2f __attribute__((ext_vector_type(2)));
typedef float v8f __attribute__((ext_vector_type(8)));

// ---------------------------------------------------------------- utilities
__global__ void zero_f32(float* __restrict__ p, long n) {
    long t = (long)blockIdx.x * blockDim.x + threadIdx.x;
    if (t < n) p[t] = 0.0f;
}

__global__ void deg_count(const int* __restrict__ idx, float* __restrict__ deg, int e) {
    int t = blockIdx.x * blockDim.x + threadIdx.x;
    if (t < e) atomicAdd(&deg[idx[t]], 1.0f);
}

__global__ void inv_sqrt_deg(float* __restrict__ d, int n) {
    int t = blockIdx.x * blockDim.x + threadIdx.x;
    if (t < n) {
        float x = d[t];
        x = x < 1.0f ? 1.0f : x;
        d[t] = 1.0f / sqrtf(x);
    }
}

// ------------------------------------------------- fp32 WMMA scaled GEMM
// X[M,128] = (A[M,K] * scale[M,None]) @ W[K,128]
// One wave per 16-row strip; holds all 8 16x16 output tiles (64 accum VGPRs).
// Per K-step of 4: one A-frag load+scale, then 8 V_WMMA_F32_16X16X4_F32.
// A-frag (16x4 f32, ISA 7.12.2): lanes 0-15 -> M=lane, K={k,k+1};
//                                lanes 16-31 -> M=lane-16, K={k+2,k+3}.
// C/D frag: VGPR r: lanes 0-15 -> M=r, lanes 16-31 -> M=r+8, N=lane%16.
__global__ void gemm_scaled_wmma(const float* __restrict__ A,
                                 const float* __restrict__ scale,
                                 const float* __restrict__ W,
                                 float* __restrict__ X,
                                 int M, int K) {
    int gw = (blockIdx.x * blockDim.x + threadIdx.x) >> 5;   // wave id = row strip
    int nstrips = M >> 4;
    if (gw >= nstrips) return;                               // wave-uniform guard

    int lane = threadIdx.x & 31;
    int m16  = lane & 15;
    int kp   = (lane >> 4) << 1;                             // 0 or 2
    int row  = (gw << 4) + m16;

    float s = scale[row];
    const float* arow = A + (size_t)row * K;

    v8f c[8];
#pragma unroll
    for (int tn = 0; tn < 8; ++tn) c[tn] = (v8f){};

    for (int k = 0; k < K; k += 4) {
        v2f a;
        a[0] = arow[k + kp] * s;
        a[1] = arow[k + kp + 1] * s;
        const float* w0 = W + (size_t)(k + kp) * CC + m16;
        const float* w1 = w0 + CC;
#pragma unroll
        for (int tn = 0; tn < 8; ++tn) {
            v2f b;
            b[0] = w0[tn * 16];
            b[1] = w1[tn * 16];
            c[tn] = __builtin_amdgcn_wmma_f32_16x16x4_f32(
                        /*neg_a=*/false, a, /*neg_b=*/false, b,
                        /*c_mod=*/(short)0, c[tn],
                        /*reuse_a=*/false, /*reuse_b=*/false);
        }
    }

    int rbase = (gw << 4) + ((lane >> 4) << 3);
#pragma unroll
    for (int tn = 0; tn < 8; ++tn) {
#pragma unroll
        for (int r = 0; r < 8; ++r)
            X[(size_t)(rbase + r) * CC + tn * 16 + m16] = c[tn][r];
    }
}

// ---------------------------------------------- edge scatter (atomic adds)
// agg[dst[e], :] += x[src[e], :]; one wave per edge, 4 channels per lane.
// x and agg are each 51 MB -> fully L2-resident (192 MB), so this runs at
// L2 rather than HBM rates.
__global__ void scatter_add_edges(const float* __restrict__ x,
                                  const int* __restrict__ src,
                                  const int* __restrict__ dst,
                                  float* __restrict__ agg, int e) {
    int g  = blockIdx.x * blockDim.x + threadIdx.x;
    int ed = g >> 5;
    if (ed >= e) return;
    int lane = g & 31;
    int s = src[ed], d = dst[ed];
    const float4 v = *reinterpret_cast<const float4*>(x + (size_t)s * CC + lane * 4);
    float* o = agg + (size_t)d * CC + lane * 4;
    atomicAdd(o + 0, v.x);
    atomicAdd(o + 1, v.y);
    atomicAdd(o + 2, v.z);
    atomicAdd(o + 3, v.w);
}

// -------------------------------- fused dst-norm + bias + LayerNorm + ELU
__global__ void finish_ln_elu(const float* __restrict__ agg,
                              const float* __restrict__ cd,
                              const float* __restrict__ bias,
                              const float* __restrict__ g,
                              const float* __restrict__ bln,
                              float* __restrict__ h) {
    __shared__ float sh[CC];
    int row = blockIdx.x;
    int t   = threadIdx.x;
    float v = agg[(size_t)row * CC + t] * cd[row] + bias[t];

    sh[t] = v; __syncthreads();
    for (int o = CC / 2; o > 0; o >>= 1) { if (t < o) sh[t] += sh[t + o]; __syncthreads(); }
    float mu = sh[0] * (1.0f / CC); __syncthreads();

    float dv = v - mu;
    sh[t] = dv * dv; __syncthreads();
    for (int o = CC / 2; o > 0; o >>= 1) { if (t < o) sh[t] += sh[t + o]; __syncthreads(); }
    float var = sh[0] * (1.0f / CC);

    float y = dv * rsqrtf(var + 1e-5f) * g[t] + bln[t];
    h[(size_t)row * CC + t] = y > 0.0f ? y : (expf(y) - 1.0f);
}

// ------------------------------------------------------- readout column mean
__global__ void colmean_acc(const float* __restrict__ h, float* __restrict__ acc,
                            int n, float inv) {
    int t = threadIdx.x;  // 128
    float s = 0.0f;
    for (int r = blockIdx.x; r < n; r += gridDim.x)
        s += h[(size_t)r * CC + t];
    atomicAdd(&acc[t], s * inv);
}

// ----------------------------------------------------------- MLP heads (tiny)
__global__ void head_mlp(const float* __restrict__ acc,
                         const float* Wm1, const float* bm1,
                         const float* Wm2, const float* bm2,
                         const float* Wd1, const float* bd1,
                         const float* Wd2, const float* bd2,
                         const float* Wd3, const float* bd3,
                         const float* Wp1, const float* bp1,
                         const float* Wp2, const float* bp2,
                         const float* Wp3, const float* bp3,
                         float* __restrict__ out) {
    __shared__ float hg[128], t1[64], zz[64], d1[128], d2[128], p1[64], p2[32];
    int t = threadIdx.x;  // 128 threads
    hg[t] = acc[t];
    __syncthreads();

    if (t < 64) { float s = bm1[t]; for (int k = 0; k < 128; ++k) s += hg[k] * Wm1[k * 64 + t]; t1[t] = fmaxf(s, 0.0f); }
    __syncthreads();
    if (t < 64) { float s = bm2[t]; for (int k = 0; k < 64; ++k) s += t1[k] * Wm2[k * 64 + t]; zz[t] = s; out[65 + t] = s; }
    __syncthreads();
    { float s = bd1[t]; for (int k = 0; k < 64; ++k) s += zz[k] * Wd1[k * 128 + t]; d1[t] = fmaxf(s, 0.0f); }
    __syncthreads();
    { float s = bd2[t]; for (int k = 0; k < 128; ++k) s += d1[k] * Wd2[k * 128 + t]; d2[t] = fmaxf(s, 0.0f); }
    __syncthreads();
    if (t < 64) { float s = bd3[t]; for (int k = 0; k < 128; ++k) s += d2[k] * Wd3[k * 64 + t]; out[t] = s; }
    if (t < 64) { float s = bp1[t]; for (int k = 0; k < 64; ++k) s += zz[k] * Wp1[k * 64 + t]; p1[t] = fmaxf(s, 0.0f); }
    __syncthreads();
    if (t < 32) { float s = bp2[t]; for (int k = 0; k < 64; ++k) s += p1[k] * Wp2[k * 32 + t]; p2[t] = fmaxf(s, 0.0f); }
    __syncthreads();
    if (t == 0) { float s = bp3[0]; for (int k = 0; k < 32; ++k) s += p2[k] * Wp3[k]; out[64] = s; }
}

// ============================================================================
extern "C" void kernel_launch(void* const* d_in, const int* in_sizes, int n_in,
                              void* d_out, int out_size, void* d_ws, size_t ws_size,
                              hipStream_t stream) {
    const float* feat_n  = (const float*)d_in[0];
    const float* feat_l  = (const float*)d_in[1];
    const int*   src_l2n = (const int*)d_in[2];
    const int*   dst_l2n = (const int*)d_in[3];
    const int*   src_n2l = (const int*)d_in[4];
    const int*   dst_n2l = (const int*)d_in[5];
    const float* W0_l2n  = (const float*)d_in[6];
    const float* b0_l2n  = (const float*)d_in[7];
    const float* W0_n2l  = (const float*)d_in[8];
    const float* b0_n2l  = (const float*)d_in[9];
    const float* W_l2n   = (const float*)d_in[10];
    const float* b_l2n   = (const float*)d_in[11];
    const float* W_n2l   = (const float*)d_in[12];
    const float* b_n2l   = (const float*)d_in[13];
    const float* ln_g_n  = (const float*)d_in[14];
    const float* ln_b_n  = (const float*)d_in[15];
    const float* ln_g_l  = (const float*)d_in[16];
    const float* ln_b_l  = (const float*)d_in[17];
    float* out = (float*)d_out;

    // ---- workspace layout (floats) ----
    float* wsf     = (float*)d_ws;
    float* cs_l2n  = wsf + 0;        // [100000]
    float* cd_l2n  = wsf + 100000;   // [100000]
    float* cs_n2l  = wsf + 200000;   // [100000]
    float* cd_n2l  = wsf + 300000;   // [100000]
    float* acc     = wsf + 400000;   // [128]
    const size_t H = (size_t)100000 * CC;  // 12.8M floats
    float* h_n   = wsf + 400128;
    float* h_l   = h_n + H;
    float* x_l   = h_l + H;
    float* x_n   = x_l + H;
    float* agg_n = x_n + H;
    float* agg_l = agg_n + H;

    // ---- degrees + readout accumulator: zero then count ----
    zero_f32<<<(400128 + 255) / 256, 256, 0, stream>>>(wsf, 400128);
    int degBlocks = (NE + 255) / 256;
    deg_count<<<degBlocks, 256, 0, stream>>>(src_l2n, cs_l2n, NE);
    deg_count<<<degBlocks, 256, 0, stream>>>(dst_l2n, cd_l2n, NE);
    deg_count<<<degBlocks, 256, 0, stream>>>(src_n2l, cs_n2l, NE);
    deg_count<<<degBlocks, 256, 0, stream>>>(dst_n2l, cd_n2l, NE);
    inv_sqrt_deg<<<(400000 + 255) / 256, 256, 0, stream>>>(wsf, 400000);

    // ---- 3 hetero-conv layers ----
    const int nstrips    = 100000 / 16;                        // 6250 waves
    const int gemmBlocks = (nstrips * 32 + 127) / 128;         // 1563 blocks (4 waves ea)
    const int scatBlocks = (int)(((long)NE * 32 + 255) / 256); // 200000
    const long aggN      = 2 * (long)H;

    for (int i = 0; i < 3; ++i) {
        const float *Aln, *Anl, *Wln, *bln, *Wnl, *bnl;
        int Kln, Knl;
        if (i == 0) {
            Aln = feat_l; Kln = DL; Wln = W0_l2n; bln = b0_l2n;
            Anl = feat_n; Knl = DN; Wnl = W0_n2l; bnl = b0_n2l;
        } else {
            Aln = h_l; Kln = CC; Wln = W_l2n + (size_t)(i - 1) * CC * CC; bln = b_l2n + (i - 1) * CC;
            Anl = h_n; Knl = CC; Wnl = W_n2l + (size_t)(i - 1) * CC * CC; bnl = b_n2l + (i - 1) * CC;
        }
        // src-norm + project (fp32 WMMA, one wave per 16x128 strip)
        gemm_scaled_wmma<<<gemmBlocks, 128, 0, stream>>>(Aln, cs_l2n, Wln, x_l, NL, Kln);
        gemm_scaled_wmma<<<gemmBlocks, 128, 0, stream>>>(Anl, cs_n2l, Wnl, x_n, NN, Knl);
        // gather + sum over edges
        zero_f32<<<(aggN + 255) / 256, 256, 0, stream>>>(agg_n, aggN);  // agg_n & agg_l contiguous
        scatter_add_edges<<<scatBlocks, 256, 0, stream>>>(x_l, src_l2n, dst_l2n, agg_n, NE);
        scatter_add_edges<<<scatBlocks, 256, 0, stream>>>(x_n, src_n2l, dst_n2l, agg_l, NE);
        // dst-norm + bias + LayerNorm + ELU
        finish_ln_elu<<<NN, CC, 0, stream>>>(agg_n, cd_l2n, bln, ln_g_n + i * CC, ln_b_n + i * CC, h_n);
        finish_ln_elu<<<NL, CC, 0, stream>>>(agg_l, cd_n2l, bnl, ln_g_l + i * CC, ln_b_l + i * CC, h_l);
    }

    // ---- readout: mean over both node types into acc[128] ----
    colmean_acc<<<512, CC, 0, stream>>>(h_n, acc, NN, 1.0f / NN);
    colmean_acc<<<512, CC, 0, stream>>>(h_l, acc, NL, 1.0f / NL);

    // ---- heads ----
    head_mlp<<<1, 128, 0, stream>>>(acc,
        (const float*)d_in[18], (const float*)d_in[19],
        (const float*)d_in[20], (const float*)d_in[21],
        (const float*)d_in[22], (const float*)d_in[23],
        (const float*)d_in[24], (const float*)d_in[25],
        (const float*)d_in[26], (const float*)d_in[27],
        (const float*)d_in[28], (const float*)d_in[29],
        (const float*)d_in[30], (const float*)d_in[31],
        (const float*)d_in[32], (const float*)d_in[33],
        out);
    (void)in_sizes; (void)n_in; (void)out_size; (void)ws_size;
}